// GNNModel_52673478918854
// MI455X (gfx1250) — compile-verified
//
#include <hip/hip_runtime.h>
#include <hip/hip_bf16.h>

typedef float v2f __attribute__((ext_vector_type(2)));
typedef float v8f __attribute__((ext_vector_type(8)));

#define NN 20000
#define EE 640000
#define FIN 128
#define HH 4
#define DD 64
#define HDIM 256
#define GG 16
#define NEG_SLOPE 0.2f

// ---------- helpers ----------
__device__ __forceinline__ unsigned f2ord(float f) {
    unsigned u = __float_as_uint(f);
    return (u & 0x80000000u) ? ~u : (u | 0x80000000u);
}
__device__ __forceinline__ float ord2f(unsigned u) {
    return (u & 0x80000000u) ? __uint_as_float(u & 0x7FFFFFFFu)
                             : __uint_as_float(~u);
}

__global__ void zero_f32(float* __restrict__ p, int n) {
    int i = blockIdx.x * blockDim.x + threadIdx.x;
    if (i < n) p[i] = 0.0f;
}
__global__ void zero_u32(unsigned* __restrict__ p, int n) {
    int i = blockIdx.x * blockDim.x + threadIdx.x;
    if (i < n) p[i] = 0u;
}

// ---------- fp32 WMMA GEMM:  Out[N,256] = X[N,K] @ W[K,256] ----------
// grid = (N/16, 256/64), block = 32; one wave computes a 16x64 strip
// (4 accumulators reuse the A fragment 4x per load).
__global__ __launch_bounds__(32)
void gemm_wmma_f32(const float* __restrict__ X, const float* __restrict__ W,
                   float* __restrict__ Out, int K) {
    const int row0 = blockIdx.x * 16;
    const int col0 = blockIdx.y * 64;
    const int lane = threadIdx.x;
    const int m    = lane & 15;    // row (A) / col (B,D) within tile
    const int half = lane >> 4;    // K-pair select for A/B halves

    v8f acc0 = {}, acc1 = {}, acc2 = {}, acc3 = {};
    const float* __restrict__ xrow = X + (size_t)(row0 + m) * K;
    const float* __restrict__ wcol = W + col0 + m;

    for (int kc = 0; kc < K; kc += 4) {
        const int kb = kc + half * 2;
        v2f a;  // A 16x4: lanes 0-15 K={0,1}, lanes 16-31 K={2,3}
        a.x = xrow[kb];
        a.y = xrow[kb + 1];
        const float* __restrict__ w0 = wcol + (size_t)kb * HDIM;
        const float* __restrict__ w1 = w0 + HDIM;
        v2f b0, b1, b2, b3;  // B 4x16: lane = N, VGPR j = K = kb + j
        b0.x = w0[0];  b0.y = w1[0];
        b1.x = w0[16]; b1.y = w1[16];
        b2.x = w0[32]; b2.y = w1[32];
        b3.x = w0[48]; b3.y = w1[48];
        acc0 = __builtin_amdgcn_wmma_f32_16x16x4_f32(false, a, false, b0, (short)0, acc0, false, false);
        acc1 = __builtin_amdgcn_wmma_f32_16x16x4_f32(false, a, false, b1, (short)0, acc1, false, false);
        acc2 = __builtin_amdgcn_wmma_f32_16x16x4_f32(false, a, false, b2, (short)0, acc2, false, false);
        acc3 = __builtin_amdgcn_wmma_f32_16x16x4_f32(false, a, false, b3, (short)0, acc3, false, false);
    }
    // D 16x16: VGPR i -> rows i (lanes 0-15) / i+8 (lanes 16-31)
#pragma unroll
    for (int i = 0; i < 8; ++i) {
        float* __restrict__ orow = Out + (size_t)(row0 + i + half * 8) * HDIM + col0 + m;
        orow[0]  = acc0[i];
        orow[16] = acc1[i];
        orow[32] = acc2[i];
        orow[48] = acc3[i];
    }
}

// ---------- attention scores: als/ald [N,H] ----------
// grid = N, block = 128 (wave per head), wave32 shuffle reduce
__global__ __launch_bounds__(128)
void attn_scores(const float* __restrict__ Hf, const float* __restrict__ a_s,
                 const float* __restrict__ a_d, float* __restrict__ als,
                 float* __restrict__ ald) {
    const int n    = blockIdx.x;
    const int head = threadIdx.x >> 5;
    const int lane = threadIdx.x & 31;
    const float* hp  = Hf  + (size_t)n * HDIM + head * DD;
    const float* asp = a_s + head * DD;
    const float* adp = a_d + head * DD;
    const float h0 = hp[lane], h1 = hp[lane + 32];
    float vs = h0 * asp[lane] + h1 * asp[lane + 32];
    float vd = h0 * adp[lane] + h1 * adp[lane + 32];
#pragma unroll
    for (int off = 16; off > 0; off >>= 1) {
        vs += __shfl_xor(vs, off, 32);
        vd += __shfl_xor(vd, off, 32);
    }
    if (lane == 0) {
        als[n * HH + head] = vs;
        ald[n * HH + head] = vd;
    }
}

// ---------- edge pass 1: segment max via ordered-uint atomicMax ----------
__global__ void edge_max(const int* __restrict__ ei, const float* __restrict__ als,
                         const float* __restrict__ ald, unsigned* __restrict__ m_ord) {
    const int t = blockIdx.x * blockDim.x + threadIdx.x;
    if (t >= EE + NN) return;
    int s, d;
    if (t < EE) { s = ei[t]; d = ei[EE + t]; } else { s = d = t - EE; }
    const float4 sv = *(const float4*)(als + (size_t)s * HH);  // b128
    const float4 dv = *(const float4*)(ald + (size_t)d * HH);
    float e[HH] = {sv.x + dv.x, sv.y + dv.y, sv.z + dv.z, sv.w + dv.w};
#pragma unroll
    for (int h = 0; h < HH; ++h) {
        const float lr = e[h] > 0.0f ? e[h] : NEG_SLOPE * e[h];  // leaky relu
        atomicMax(&m_ord[d * HH + h], f2ord(lr));
    }
}

// ---------- edge pass 2: w = exp(e - max), denom += w ----------
__global__ void edge_w(const int* __restrict__ ei, const float* __restrict__ als,
                       const float* __restrict__ ald, const unsigned* __restrict__ m_ord,
                       float* __restrict__ w_edge, float* __restrict__ denom) {
    const int t = blockIdx.x * blockDim.x + threadIdx.x;
    if (t >= EE + NN) return;
    int s, d;
    if (t < EE) { s = ei[t]; d = ei[EE + t]; } else { s = d = t - EE; }
    const float4 sv = *(const float4*)(als + (size_t)s * HH);
    const float4 dv = *(const float4*)(ald + (size_t)d * HH);
    const uint4  mv = *(const uint4*)(m_ord + (size_t)d * HH);
    float e[HH] = {sv.x + dv.x, sv.y + dv.y, sv.z + dv.z, sv.w + dv.w};
    const unsigned mo[HH] = {mv.x, mv.y, mv.z, mv.w};
    float4 wv;
    float* wp = &wv.x;
#pragma unroll
    for (int h = 0; h < HH; ++h) {
        const float lr = e[h] > 0.0f ? e[h] : NEG_SLOPE * e[h];
        const float w  = __expf(lr - ord2f(mo[h]));
        wp[h] = w;
        atomicAdd(&denom[d * HH + h], w);
    }
    *(float4*)(w_edge + (size_t)t * HH) = wv;  // b128 store
}

// ---------- edge pass 3: out[dst] += h[src] * alpha  (wave per edge) ----------
__global__ __launch_bounds__(256)
void edge_aggr(const int* __restrict__ ei, const float* __restrict__ Hf,
               const float* __restrict__ w_edge, const float* __restrict__ denom,
               float* __restrict__ Outc) {
    const int wave = threadIdx.x >> 5;
    const int lane = threadIdx.x & 31;
    const int e    = blockIdx.x * 8 + wave;
    if (e >= EE + NN) return;
    int s, d;
    if (e < EE) { s = ei[e]; d = ei[EE + e]; } else { s = d = e - EE; }
    const float4 wv = *(const float4*)(w_edge + (size_t)e * HH);
    const float4 dn = *(const float4*)(denom  + (size_t)d * HH);
    const float alpha[HH] = {wv.x / (dn.x + 1e-16f), wv.y / (dn.y + 1e-16f),
                             wv.z / (dn.z + 1e-16f), wv.w / (dn.w + 1e-16f)};
    const float* __restrict__ hs = Hf   + (size_t)s * HDIM;
    float* __restrict__       od = Outc + (size_t)d * HDIM;
#pragma unroll
    for (int i = 0; i < 2; ++i) {
        const int base = i * 128 + lane * 4;            // 4-chunk stays in one head
        const float4 hv = *(const float4*)(hs + base);  // b128 gather
        const float  al = alpha[base >> 6];
        atomicAdd(&od[base + 0], hv.x * al);
        atomicAdd(&od[base + 1], hv.y * al);
        atomicAdd(&od[base + 2], hv.z * al);
        atomicAdd(&od[base + 3], hv.w * al);
    }
}

// ---------- bias + ELU ----------
__global__ void bias_elu(const float* __restrict__ C, const float* __restrict__ b,
                         float* __restrict__ Xo, int n) {
    const int i = blockIdx.x * blockDim.x + threadIdx.x;
    if (i >= n) return;
    const float v = C[i] + b[i & (HDIM - 1)];
    Xo[i] = v > 0.0f ? v : (__expf(v) - 1.0f);
}

// ---------- global mean pool ----------
__global__ void pool_sum(const float* __restrict__ Xf, const int* __restrict__ batch,
                         float* __restrict__ gsum) {
    const int i = blockIdx.x * blockDim.x + threadIdx.x;
    if (i >= NN * HDIM) return;
    const int node = i >> 8;
    atomicAdd(&gsum[batch[node] * HDIM + (i & (HDIM - 1))], Xf[i]);
}
__global__ void pool_cnt(const int* __restrict__ batch, float* __restrict__ cnt) {
    const int i = blockIdx.x * blockDim.x + threadIdx.x;
    if (i < NN) atomicAdd(&cnt[batch[i]], 1.0f);
}

// ---------- MLP head: 256 -> 64 (BN, relu) -> 32 (relu) -> 1 ----------
__global__ __launch_bounds__(64)
void mlp_head(const float* __restrict__ gsum, const float* __restrict__ cnt,
              const float* __restrict__ fc1w, const float* __restrict__ fc1b,
              const float* __restrict__ fc2w, const float* __restrict__ fc2b,
              const float* __restrict__ fc3w, const float* __restrict__ fc3b,
              const float* __restrict__ gmm, const float* __restrict__ gbb,
              const float* __restrict__ bmean, const float* __restrict__ bvar,
              float* __restrict__ out) {
    __shared__ float gmean[HDIM];
    __shared__ float t1[64];
    __shared__ float t2[32];
    const int g = blockIdx.x;
    const int t = threadIdx.x;
    float c = cnt[g];
    c = c > 1.0f ? c : 1.0f;
    for (int k = t; k < HDIM; k += 64) gmean[k] = gsum[g * HDIM + k] / c;
    __syncthreads();
    float a1 = fc1b[t];
    for (int k = 0; k < HDIM; ++k) a1 += gmean[k] * fc1w[k * 64 + t];
    a1 = (a1 - bmean[t]) * rsqrtf(bvar[t] + 1e-5f) * gmm[t] + gbb[t];
    t1[t] = a1 > 0.0f ? a1 : 0.0f;
    __syncthreads();
    if (t < 32) {
        float a2 = fc2b[t];
        for (int k = 0; k < 64; ++k) a2 += t1[k] * fc2w[k * 32 + t];
        t2[t] = a2 > 0.0f ? a2 : 0.0f;
    }
    __syncthreads();
    if (t == 0) {
        float a3 = fc3b[0];
        for (int k = 0; k < 32; ++k) a3 += t2[k] * fc3w[k];
        out[g] = a3;
    }
}

extern "C" void kernel_launch(void* const* d_in, const int* in_sizes, int n_in,
                              void* d_out, int out_size, void* d_ws, size_t ws_size,
                              hipStream_t stream) {
    const float* x    = (const float*)d_in[0];
    const int*   ei   = (const int*)d_in[1];
    const int*   batch= (const int*)d_in[2];
    const float* W1   = (const float*)d_in[3];
    const float* as1  = (const float*)d_in[4];
    const float* ad1  = (const float*)d_in[5];
    const float* b1   = (const float*)d_in[6];
    const float* W2   = (const float*)d_in[7];
    const float* as2  = (const float*)d_in[8];
    const float* ad2  = (const float*)d_in[9];
    const float* b2   = (const float*)d_in[10];
    const float* W3   = (const float*)d_in[11];
    const float* as3  = (const float*)d_in[12];
    const float* ad3  = (const float*)d_in[13];
    const float* b3   = (const float*)d_in[14];
    const float* fc1w = (const float*)d_in[15];
    const float* fc1b = (const float*)d_in[16];
    const float* fc2w = (const float*)d_in[17];
    const float* fc2b = (const float*)d_in[18];
    const float* fc3w = (const float*)d_in[19];
    const float* fc3b = (const float*)d_in[20];
    const float* bng  = (const float*)d_in[21];
    const float* bnb  = (const float*)d_in[22];
    const float* bnm  = (const float*)d_in[23];
    const float* bnv  = (const float*)d_in[24];
    float* out = (float*)d_out;

    const int EP = EE + NN;

    // workspace carve-up (all fp32/u32; every region 16B aligned)
    float*    bufH  = (float*)d_ws;                      // [N, 256] gemm result
    float*    bufC  = bufH  + (size_t)NN * HDIM;         // [N, 256] aggregated
    float*    bufX  = bufC  + (size_t)NN * HDIM;         // [N, 256] activated
    float*    als   = bufX  + (size_t)NN * HDIM;         // [N, H]
    float*    ald   = als   + (size_t)NN * HH;           // [N, H]
    unsigned* m_ord = (unsigned*)(ald + (size_t)NN * HH);// [N, H]
    float*    denom = (float*)(m_ord + (size_t)NN * HH); // [N, H]
    float*    wedge = denom + (size_t)NN * HH;           // [E+N, H]
    float*    gsum  = wedge + (size_t)EP * HH;           // [G, 256]
    float*    cnt   = gsum  + (size_t)GG * HDIM;         // [G]

    const int nh  = NN * HH;
    const int nhd = NN * HDIM;

    auto layer = [&](const float* Xin, int K, const float* W, const float* a_s,
                     const float* a_d, const float* bias) {
        dim3 ggrid(NN / 16, HDIM / 64);
        gemm_wmma_f32<<<ggrid, 32, 0, stream>>>(Xin, W, bufH, K);
        attn_scores<<<NN, 128, 0, stream>>>(bufH, a_s, a_d, als, ald);
        zero_u32<<<(nh + 255) / 256, 256, 0, stream>>>(m_ord, nh);
        zero_f32<<<(nh + 255) / 256, 256, 0, stream>>>(denom, nh);
        zero_f32<<<(nhd + 255) / 256, 256, 0, stream>>>(bufC, nhd);
        edge_max<<<(EP + 255) / 256, 256, 0, stream>>>(ei, als, ald, m_ord);
        edge_w<<<(EP + 255) / 256, 256, 0, stream>>>(ei, als, ald, m_ord, wedge, denom);
        edge_aggr<<<(EP + 7) / 8, 256, 0, stream>>>(ei, bufH, wedge, denom, bufC);
        bias_elu<<<(nhd + 255) / 256, 256, 0, stream>>>(bufC, bias, bufX, nhd);
    };

    layer(x,    FIN,  W1, as1, ad1, b1);
    layer(bufX, HDIM, W2, as2, ad2, b2);
    layer(bufX, HDIM, W3, as3, ad3, b3);

    zero_f32<<<(GG * HDIM + 255) / 256, 256, 0, stream>>>(gsum, GG * HDIM);
    zero_f32<<<1, 256, 0, stream>>>(cnt, GG);
    pool_sum<<<(nhd + 255) / 256, 256, 0, stream>>>(bufX, batch, gsum);
    pool_cnt<<<(NN + 255) / 256, 256, 0, stream>>>(batch, cnt);
    mlp_head<<<GG, 64, 0, stream>>>(gsum, cnt, fc1w, fc1b, fc2w, fc2b, fc3w, fc3b,
                                    bng, bnb, bnm, bnv, out);
}